// Model_36550171689393
// MI455X (gfx1250) — compile-verified
//
#include <hip/hip_runtime.h>
#include <math.h>

// Problem constants (from reference)
#define V_   100000
#define EMB_ 300
#define H_   64
#define N_   20000
#define E_   160000
#define B_   4096
#define L_   2
#define T_   3

typedef float v2f __attribute__((ext_vector_type(2)));
typedef float v8f __attribute__((ext_vector_type(8)));
typedef unsigned int u32x4 __attribute__((ext_vector_type(4)));
typedef int i32x4 __attribute__((ext_vector_type(4)));
typedef int i32x8 __attribute__((ext_vector_type(8)));

// ---------------------------------------------------------------------------
// WMMA: D = A(16x4, f32) * B(4x16, f32) + C(16x16, f32)   [codegen-verified]
// A frag: lane&15 = M row; lanes16-31 hold K+2/K+3. B frag symmetric over N.
// C frag: col = lane&15, row = v + (lane>=16 ? 8 : 0).
// ---------------------------------------------------------------------------
static __device__ __forceinline__ v8f wmma_f32_16x16x4(v2f a, v2f b, v8f c) {
  return __builtin_amdgcn_wmma_f32_16x16x4_f32(false, a, false, b, (short)0, c,
                                               false, false);
}

// ---------------------------------------------------------------------------
// Hardware f32 atomic add (VGLOBAL GLOBAL_ATOMIC_ADD_F32, no return)
// ---------------------------------------------------------------------------
static __device__ __forceinline__ void atomic_add_f32(float* dst, float val) {
  asm volatile("global_atomic_add_f32 %0, %1, off"
               :: "v"(dst), "v"(val)
               : "memory");
}

// ---------------------------------------------------------------------------
// TDM: async 2-D f32 tile copy  global -> LDS  (one descriptor, count=1)
// D# group0: count=1 | lds_addr | global_addr | type=2
// D# group1: data_size=4B | tensor_dim0/1 | tile_dim0/1 (=full) | stride0
// ---------------------------------------------------------------------------
static __device__ __forceinline__ void tdm_load_2d_f32(unsigned lds_byte_addr,
                                                       const void* gptr,
                                                       unsigned dim0,
                                                       unsigned dim1) {
#if __has_builtin(__builtin_amdgcn_tensor_load_to_lds)
  unsigned long long ga = (unsigned long long)(uintptr_t)gptr;
  u32x4 g0;
  g0[0] = 1u;                                   // count=1, flags 0, no gather
  g0[1] = lds_byte_addr;                        // bits 63:32  lds_addr
  g0[2] = (unsigned)(ga & 0xFFFFFFFFu);         // global_addr[31:0]
  g0[3] = (unsigned)((ga >> 32) & 0x1FFFFFFu)   // global_addr[56:32]
          | (2u << 30);                         // type=2 ("image")
  i32x8 g1;
  g1[0] = (int)(2u << 16);                      // data_size=2 -> 4 bytes
  g1[1] = (int)((dim0 & 0xFFFFu) << 16);        // tensor_dim0[15:0]
  g1[2] = (int)(((dim0 >> 16) & 0xFFFFu) | ((dim1 & 0xFFFFu) << 16));
  g1[3] = (int)(((dim1 >> 16) & 0xFFFFu) | ((dim0 & 0xFFFFu) << 16)); // tile_dim0
  g1[4] = (int)(dim1 & 0xFFFFu);                // tile_dim1 ; tile_dim2=0
  g1[5] = (int)dim0;                            // tensor_dim0_stride[31:0]
  g1[6] = 0;                                    // stride0[47:32], stride1[15:0]
  g1[7] = 0;                                    // stride1[47:16]
  i32x4 z4 = {0, 0, 0, 0};
#if __has_include(<hip/amd_detail/amd_gfx1250_TDM.h>)
  i32x8 z8 = {0, 0, 0, 0, 0, 0, 0, 0};
  __builtin_amdgcn_tensor_load_to_lds(g0, g1, z4, z4, z8, 0);   // clang-23 form
#else
  __builtin_amdgcn_tensor_load_to_lds(g0, g1, z4, z4, 0);       // clang-22 form
#endif
#if __has_builtin(__builtin_amdgcn_s_wait_tensorcnt)
  __builtin_amdgcn_s_wait_tensorcnt(0);
#else
  asm volatile("s_wait_tensorcnt 0x0" ::: "memory");
#endif
#else
  (void)lds_byte_addr; (void)gptr; (void)dim0; (void)dim1;
#endif
}

// ---------------------------------------------------------------------------
// Zero a float buffer
// ---------------------------------------------------------------------------
__global__ void k_zero(float* __restrict__ p, int n) {
  int i = blockIdx.x * blockDim.x + threadIdx.x;
  if (i < n) p[i] = 0.0f;
}

// ---------------------------------------------------------------------------
// Gather embedding rows + project:  H[r,:] = emb[ind[r],:] @ W(300x64) + b
// Grid: 2N/16 blocks x 128 threads (4 waves; wave w -> col tile w)
// ---------------------------------------------------------------------------
__global__ void k_gather_project(const int* __restrict__ ind0,
                                 const int* __restrict__ ind1,
                                 const float* __restrict__ emb,
                                 const float* __restrict__ W,
                                 const float* __restrict__ bias,
                                 float* __restrict__ Hout) {
  const int tile = blockIdx.x;                 // 16 rows of [0, 2N)
  const int wave = threadIdx.x >> 5;           // 0..3
  const int lane = threadIdx.x & 31;
  const int r16  = lane & 15;
  const int kh   = lane >> 4;
  const int ko   = kh * 2;

  const int rbase  = tile * 16;
  const int branch = rbase / N_;               // uniform per tile (16 | N)
  const int local  = rbase - branch * N_ + r16;
  const int ind    = branch ? ind1[local] : ind0[local];
  const float* arow = emb + (long)ind * EMB_;  // gathered A row (this lane)
  const int col = wave * 16 + r16;             // B/C column (this lane)

  v8f c = {0.f, 0.f, 0.f, 0.f, 0.f, 0.f, 0.f, 0.f};
  for (int k = 0; k < EMB_; k += 4) {
    v2f a = *(const v2f*)(arow + k + ko);      // 8B aligned: ind*300 even
    v2f b;
    b.x = W[(k + ko) * H_ + col];
    b.y = W[(k + ko + 1) * H_ + col];
    c = wmma_f32_16x16x4(a, b, c);
  }

  const float bv = bias[col];
  float* out = Hout + (long)rbase * H_ + wave * 16;
#pragma unroll
  for (int v = 0; v < 8; ++v) {
    const int rr = v + kh * 8;
    out[rr * H_ + r16] = c[v] + bv;
  }
}

// ---------------------------------------------------------------------------
// Fused message GEMM + scatter:  INC[tgt[e],:] += h[src[e],:] @ Wm + bm
// Wm staged into LDS via the Tensor Data Mover (wave 0 issues, all consume).
// Grid: 2E/16 blocks x 128 threads (4 waves; wave w -> col tile w)
// ---------------------------------------------------------------------------
__global__ void k_message_scatter(const float* __restrict__ Hin,
                                  const int* __restrict__ adj0,
                                  const int* __restrict__ adj1,
                                  const float* __restrict__ Wm,  // 64x64
                                  const float* __restrict__ bm,  // 64
                                  float* __restrict__ INC) {
  __shared__ float s_W[H_ * H_];               // 16 KB weight stage
  __shared__ int   s_tgt[16];

  const int tile = blockIdx.x;                 // 16 edges of [0, 2E)
  const int wave = threadIdx.x >> 5;
  const int lane = threadIdx.x & 31;
  const int r16  = lane & 15;
  const int kh   = lane >> 4;
  const int ko   = kh * 2;

  const int ebase  = tile * 16;
  const int branch = ebase / E_;               // uniform per tile (16 | E)
  const int* adj   = branch ? adj1 : adj0;
  const int lebase = ebase - branch * E_;

  // Wave 0: async TDM copy of Wm -> s_W, then wait TENSORcnt==0.
  if (wave == 0) {
#if __has_builtin(__builtin_amdgcn_tensor_load_to_lds)
    tdm_load_2d_f32((unsigned)(uintptr_t)&s_W[0], Wm, H_, H_);
#else
    for (int i = (int)lane; i < H_ * H_; i += 32) s_W[i] = Wm[i];
#endif
  }
  if (threadIdx.x < 16) s_tgt[threadIdx.x] = adj[(lebase + (int)threadIdx.x) * 2 + 1];
  __syncthreads();

  const int src = adj[(lebase + r16) * 2 + 0]; // A row (this lane)
  const float* arow = Hin + (long)(branch * N_ + src) * H_;
  const int col = wave * 16 + r16;             // B/C column (this lane)

  v8f c = {0.f, 0.f, 0.f, 0.f, 0.f, 0.f, 0.f, 0.f};
#pragma unroll
  for (int k = 0; k < H_; k += 4) {
    v2f a = *(const v2f*)(arow + k + ko);
    v2f b;
    b.x = s_W[(k + ko) * H_ + col];
    b.y = s_W[(k + ko + 1) * H_ + col];
    c = wmma_f32_16x16x4(a, b, c);
  }

  const float bv = bm[col];
  const long nb = (long)branch * N_;
#pragma unroll
  for (int v = 0; v < 8; ++v) {
    const int rr = v + kh * 8;
    atomic_add_f32(INC + (nb + s_tgt[rr]) * (long)H_ + col, c[v] + bv);
  }
}

// ---------------------------------------------------------------------------
// Fused GRU cell: gi = INC @ Wih^T + bih ; gh = h @ Whh^T + bhh ;
//                 h = (1-z)*n + z*h ; INC = 0 (ready for next iteration)
// Grid: 2N/16 blocks x 384 threads (12 waves; wave w -> 16-col slice of 192)
// ---------------------------------------------------------------------------
__global__ void k_gru(float* __restrict__ Hbuf,
                      float* __restrict__ INC,
                      const float* __restrict__ Wih,   // 192x64 (row-major)
                      const float* __restrict__ Whh,   // 192x64
                      const float* __restrict__ bih,   // 192
                      const float* __restrict__ bhh) { // 192
  __shared__ float sgi[16 * 192];
  __shared__ float sgh[16 * 192];

  const int wave = threadIdx.x >> 5;           // 0..11
  const int lane = threadIdx.x & 31;
  const int r16  = lane & 15;
  const int kh   = lane >> 4;
  const int ko   = kh * 2;
  const int rbase = blockIdx.x * 16;           // rows of [0, 2N)
  const int cb    = wave * 16;                 // output-col tile base (0..176)

  const float* airow = INC  + (long)(rbase + r16) * H_;  // A rows for gi
  const float* ahrow = Hbuf + (long)(rbase + r16) * H_;  // A rows for gh
  const int colj = cb + r16;                   // this lane's B/C column j

  v8f cgi = {0.f, 0.f, 0.f, 0.f, 0.f, 0.f, 0.f, 0.f};
  v8f cgh = {0.f, 0.f, 0.f, 0.f, 0.f, 0.f, 0.f, 0.f};
#pragma unroll
  for (int k = 0; k < H_; k += 4) {
    v2f ai = *(const v2f*)(airow + k + ko);
    v2f ah = *(const v2f*)(ahrow + k + ko);
    // B[k, j] = W[j, k]  (x@W^T): consecutive k -> contiguous -> v2f load
    v2f bi = *(const v2f*)(Wih + (long)colj * H_ + k + ko);
    v2f bh = *(const v2f*)(Whh + (long)colj * H_ + k + ko);
    cgi = wmma_f32_16x16x4(ai, bi, cgi);
    cgh = wmma_f32_16x16x4(ah, bh, cgh);
  }

#pragma unroll
  for (int v = 0; v < 8; ++v) {
    const int rr = v + kh * 8;
    sgi[rr * 192 + colj] = cgi[v];
    sgh[rr * 192 + colj] = cgh[v];
  }
  __syncthreads();

  // Elementwise gates: 16 rows x 64 outputs
  for (int idx = threadIdx.x; idx < 16 * H_; idx += blockDim.x) {
    const int row = idx >> 6;
    const int j   = idx & 63;
    const float gi_r = sgi[row * 192 + j]       + bih[j];
    const float gh_r = sgh[row * 192 + j]       + bhh[j];
    const float gi_z = sgi[row * 192 + 64 + j]  + bih[64 + j];
    const float gh_z = sgh[row * 192 + 64 + j]  + bhh[64 + j];
    const float gi_n = sgi[row * 192 + 128 + j] + bih[128 + j];
    const float gh_n = sgh[row * 192 + 128 + j] + bhh[128 + j];
    const float r = 1.0f / (1.0f + __expf(-(gi_r + gh_r)));
    const float z = 1.0f / (1.0f + __expf(-(gi_z + gh_z)));
    const float n = tanhf(gi_n + r * gh_n);
    const long off = (long)(rbase + row) * H_ + j;
    const float hold = Hbuf[off];
    Hbuf[off] = (1.0f - z) * n + z * hold;
    INC[off]  = 0.0f;  // reset accumulator for next iteration
  }
}

// ---------------------------------------------------------------------------
// Classifier + per-block BCE partial sums. Grid: B/256 blocks x 256 threads.
// ---------------------------------------------------------------------------
__global__ void k_classifier(const float* __restrict__ Hbuf,
                             const int* __restrict__ prop0,
                             const int* __restrict__ prop1,
                             const int* __restrict__ labels,
                             const float* __restrict__ W1,   // 128x16
                             const float* __restrict__ b1,   // 16
                             const float* __restrict__ W2,   // 16
                             const float* __restrict__ b2,   // 1
                             float* __restrict__ logits,     // B
                             float* __restrict__ partials) { // gridDim.x
  __shared__ float red[256];
  const int b = blockIdx.x * blockDim.x + threadIdx.x;

  const float* g0 = Hbuf + (long)prop0[b] * H_;
  const float* g1 = Hbuf + (long)(N_ + prop1[b]) * H_;

  float acc = b2[0];
#pragma unroll
  for (int j = 0; j < 16; ++j) {
    float s = b1[j];
    for (int k = 0; k < H_; ++k) s = fmaf(g0[k], W1[k * 16 + j], s);
    for (int k = 0; k < H_; ++k) s = fmaf(g1[k], W1[(H_ + k) * 16 + j], s);
    s = fmaxf(s, 0.0f);                       // relu
    acc = fmaf(s, W2[j], acc);
  }
  const float p = 1.0f / (1.0f + __expf(-acc));
  logits[b] = p;

  const float eps = 1e-7f;
  const float y = (float)labels[b];
  const float pc = fminf(fmaxf(p, eps), 1.0f);
  const float qc = fminf(fmaxf(1.0f - p, eps), 1.0f);
  red[threadIdx.x] = y * __logf(pc) + (1.0f - y) * __logf(qc);
  __syncthreads();
  for (int s = 128; s > 0; s >>= 1) {
    if (threadIdx.x < s) red[threadIdx.x] += red[threadIdx.x + s];
    __syncthreads();
  }
  if (threadIdx.x == 0) partials[blockIdx.x] = red[0];
}

__global__ void k_loss_reduce(const float* __restrict__ partials, int nparts,
                              float* __restrict__ loss_out) {
  if (threadIdx.x == 0) {
    float s = 0.0f;
    for (int i = 0; i < nparts; ++i) s += partials[i];
    loss_out[0] = -s / (float)B_;
  }
}

// ---------------------------------------------------------------------------
// Host-side launch sequence
// ---------------------------------------------------------------------------
extern "C" void kernel_launch(void* const* d_in, const int* in_sizes, int n_in,
                              void* d_out, int out_size, void* d_ws,
                              size_t ws_size, hipStream_t stream) {
  (void)in_sizes; (void)n_in; (void)out_size; (void)ws_size;

  const int*   emb_ind0 = (const int*)d_in[0];
  const int*   emb_ind1 = (const int*)d_in[1];
  const int*   adj0     = (const int*)d_in[2];
  const int*   adj1     = (const int*)d_in[3];
  const int*   prop0    = (const int*)d_in[4];
  const int*   prop1    = (const int*)d_in[5];
  const int*   labels   = (const int*)d_in[6];
  const float* emb      = (const float*)d_in[7];
  const float* projW    = (const float*)d_in[8];
  const float* projB    = (const float*)d_in[9];
  const float* msgW     = (const float*)d_in[10];  // [L,64,64]
  const float* msgB     = (const float*)d_in[11];  // [L,64]
  const float* gruWih   = (const float*)d_in[12];  // [L,192,64]
  const float* gruWhh   = (const float*)d_in[13];  // [L,192,64]
  const float* gruBih   = (const float*)d_in[14];  // [L,192]
  const float* gruBhh   = (const float*)d_in[15];  // [L,192]
  const float* claW1    = (const float*)d_in[16];  // [128,16]
  const float* claB1    = (const float*)d_in[17];
  const float* claW2    = (const float*)d_in[18];  // [16,1]
  const float* claB2    = (const float*)d_in[19];

  float* out_logits = (float*)d_out;           // B
  float* out_loss   = out_logits + B_;         // 1

  // Workspace layout
  const long NH = (long)2 * N_ * H_;           // 2,560,000 floats
  float* Hbuf     = (float*)d_ws;
  float* INC      = Hbuf + NH;
  float* partials = INC + NH;                  // 16 floats

  // 1) zero INC accumulator (re-zeroed inside k_gru each iteration)
  {
    const int n = (int)NH;
    k_zero<<<(n + 255) / 256, 256, 0, stream>>>(INC, n);
  }

  // 2) gather + projection -> Hbuf  (both branches fused: rows 0..2N)
  k_gather_project<<<(2 * N_) / 16, 128, 0, stream>>>(
      emb_ind0, emb_ind1, emb, projW, projB, Hbuf);

  // 3) GGNN: L layers x T propagation steps
  for (int l = 0; l < L_; ++l) {
    const float* Wm  = msgW   + (long)l * H_ * H_;
    const float* bm  = msgB   + (long)l * H_;
    const float* Wih = gruWih + (long)l * 3 * H_ * H_;
    const float* Whh = gruWhh + (long)l * 3 * H_ * H_;
    const float* bih = gruBih + (long)l * 3 * H_;
    const float* bhh = gruBhh + (long)l * 3 * H_;
    for (int t = 0; t < T_; ++t) {
      k_message_scatter<<<(2 * E_) / 16, 128, 0, stream>>>(
          Hbuf, adj0, adj1, Wm, bm, INC);
      k_gru<<<(2 * N_) / 16, 384, 0, stream>>>(Hbuf, INC, Wih, Whh, bih, bhh);
    }
  }

  // 4) classifier + loss
  k_classifier<<<B_ / 256, 256, 0, stream>>>(
      Hbuf, prop0, prop1, labels, claW1, claB1, claW2, claB2,
      out_logits, partials);
  k_loss_reduce<<<1, 32, 0, stream>>>(partials, B_ / 256, out_loss);
}